// RGCNLayer_30073361007325
// MI455X (gfx1250) — compile-verified
//
#include <hip/hip_runtime.h>

// Problem constants (from reference).
#define NN   50000
#define EE   1600000
#define RR   8
#define DI   128
#define DO_  128
#define NT   (NN / 16)      // 3125 node tiles, exact
#define NRD  (RR * DO_)     // 1024: stride of one node's row block in `transformed`

typedef __attribute__((ext_vector_type(2))) float v2f;
typedef __attribute__((ext_vector_type(8))) float v8f;

// ---------------------------------------------------------------------------
// K0: zero an int array (destination-degree histogram).
// ---------------------------------------------------------------------------
__global__ __launch_bounds__(256) void k_zero(int* __restrict__ p, int n) {
    int i = blockIdx.x * 256 + threadIdx.x;
    if (i < n) p[i] = 0;
}

// ---------------------------------------------------------------------------
// K1: transformed[n, r, o] = sum_k h[n, k] * W[r, k, o] via V_WMMA_F32_16X16X4_F32.
// Block = 256 threads = 8 waves; wave w handles relation r = w.
// Grid.x = 3125 node tiles (16 nodes each, exact).
// A (16x4 f32) layout: lanes 0-15 hold M=0..15; v0 = K = 2*khalf, v1 = K+1.
// C (16x16 f32) layout: VGPR v = row v (lanes 0-15) / row v+8 (lanes 16-31),
// column = lane % 16.
// ---------------------------------------------------------------------------
__global__ __launch_bounds__(256) void k_gemm(const float* __restrict__ h,
                                              const float* __restrict__ w,
                                              float* __restrict__ t) {
    __shared__ float hA[16][DI + 4];        // +4 pad: conflict-free strided reads
    const int n0  = blockIdx.x * 16;
    const int tid = threadIdx.x;

    // Cooperative 16x128 f32 tile load (global_load_b128, fully coalesced).
    for (int i = tid; i < 16 * DI / 4; i += 256) {
        const int row = (i * 4) / DI;
        const int col = (i * 4) % DI;
        const float4 v = *(const float4*)(h + (size_t)(n0 + row) * DI + col);
        *(float4*)&hA[row][col] = v;
    }
    __syncthreads();

    const int lane  = tid & 31;
    const int r     = tid >> 5;     // 8 waves == 8 relations
    const int mrow  = lane & 15;
    const int khalf = lane >> 4;    // this half-wave supplies K = 4k+2*khalf, +1

    // Preload all A fragments for this wave (reused across all 8 output tiles).
    v2f areg[32];
#pragma unroll
    for (int ks = 0; ks < 32; ++ks) {
        const float2 a2 = *(const float2*)&hA[mrow][ks * 4 + 2 * khalf];
        areg[ks].x = a2.x;
        areg[ks].y = a2.y;
    }

    const float* wr = w + (size_t)r * DI * DO_;   // W[r, :, :], row-major [k][o]

    for (int ot = 0; ot < DO_ / 16; ++ot) {
        const int o0 = ot * 16;
        v8f c = {};
#pragma unroll
        for (int ks = 0; ks < 32; ++ks) {
            const int kb = ks * 4 + 2 * khalf;
            v2f b;
            b.x = wr[(size_t)kb * DO_ + o0 + mrow];
            b.y = wr[(size_t)(kb + 1) * DO_ + o0 + mrow];
            c = __builtin_amdgcn_wmma_f32_16x16x4_f32(
                    /*neg_a=*/false, areg[ks], /*neg_b=*/false, b,
                    /*c_mod=*/(short)0, c, /*reuse_a=*/false, /*reuse_b=*/false);
        }
        // Store the 16x16 C tile: t[(n0+row)*1024 + r*128 + o0 + col].
        const size_t colbase = ((size_t)n0 * RR + r) * DO_ + o0 + mrow;
#pragma unroll
        for (int v = 0; v < 8; ++v) {
            const int row = khalf * 8 + v;
            t[colbase + (size_t)row * NRD] = c[v];
        }
    }
}

// ---------------------------------------------------------------------------
// K2: histogram of destination degrees (int atomics only: 1.6M, cheap).
// ---------------------------------------------------------------------------
__global__ __launch_bounds__(256) void k_hist(const int* __restrict__ dst,
                                              int* __restrict__ cnt) {
    int e = blockIdx.x * 256 + threadIdx.x;
    if (e < EE) atomicAdd(&cnt[dst[e]], 1);
}

// ---------------------------------------------------------------------------
// K3: single-block exclusive scan over 50000 counts -> offsets[N+1], cursor[N].
// ---------------------------------------------------------------------------
__global__ __launch_bounds__(1024) void k_scan(const int* __restrict__ cnt,
                                               int* __restrict__ offs,
                                               int* __restrict__ cur) {
    __shared__ int sh[1024];
    __shared__ int carry_s;
    const int tid = threadIdx.x;
    if (tid == 0) carry_s = 0;
    __syncthreads();

    for (int base = 0; base < NN; base += 1024) {
        const int i = base + tid;
        const int v = (i < NN) ? cnt[i] : 0;
        int acc = v;
        sh[tid] = acc;
        __syncthreads();
        for (int d = 1; d < 1024; d <<= 1) {
            const int add = (tid >= d) ? sh[tid - d] : 0;
            __syncthreads();
            acc += add;
            sh[tid] = acc;
            __syncthreads();
        }
        const int c0 = carry_s;
        const int ex = c0 + acc - v;
        if (i < NN) { offs[i] = ex; cur[i] = ex; }
        __syncthreads();
        if (tid == 1023) carry_s = c0 + acc;   // acc == chunk total in lane 1023
        __syncthreads();
    }
    if (tid == 0) offs[NN] = carry_s;
}

// ---------------------------------------------------------------------------
// K4: scatter edges into CSR slots. src < 65536 and rel < 8 pack into one int.
// ---------------------------------------------------------------------------
__global__ __launch_bounds__(256) void k_scatter(const int* __restrict__ src,
                                                 const int* __restrict__ dst,
                                                 const int* __restrict__ rel,
                                                 const float* __restrict__ nrm,
                                                 int* __restrict__ cur,
                                                 int* __restrict__ epack,
                                                 float* __restrict__ enorm) {
    int e = blockIdx.x * 256 + threadIdx.x;
    if (e >= EE) return;
    const int d   = dst[e];
    const int pos = atomicAdd(&cur[d], 1);
    epack[pos] = src[e] | (rel[e] << 16);
    enorm[pos] = nrm[e];
}

// ---------------------------------------------------------------------------
// K5: one wave per destination node; 4 f32 accumulators per lane (128 outs),
// 512B fully-coalesced gathers from `transformed`, relu folded into the store.
// No floating-point atomics anywhere.
// ---------------------------------------------------------------------------
__global__ __launch_bounds__(256) void k_gather(const float* __restrict__ t,
                                                const int* __restrict__ offs,
                                                const int* __restrict__ epack,
                                                const float* __restrict__ enorm,
                                                float* __restrict__ out) {
    const int node = blockIdx.x * 8 + (threadIdx.x >> 5);   // 6250*8 == 50000
    const int lane = threadIdx.x & 31;
    if (node >= NN) return;
    const int beg = offs[node];
    const int end = offs[node + 1];

    float ax = 0.f, ay = 0.f, az = 0.f, aw = 0.f;
    for (int j = beg; j < end; ++j) {
        const int   p  = epack[j];     // wave-uniform
        const float nm = enorm[j];     // wave-uniform
        const size_t base = ((size_t)(p & 0xFFFF) * RR + (p >> 16)) * DO_;
        const float4 v = *(const float4*)(t + base + lane * 4);
        ax += nm * v.x; ay += nm * v.y; az += nm * v.z; aw += nm * v.w;
    }
    float4 o4;
    o4.x = fmaxf(ax, 0.f); o4.y = fmaxf(ay, 0.f);
    o4.z = fmaxf(az, 0.f); o4.w = fmaxf(aw, 0.f);
    *(float4*)(out + (size_t)node * DO_ + lane * 4) = o4;
}

// ---------------------------------------------------------------------------
// Launch wrapper.
// Workspace layout (256B-aligned regions, total ~218.2 MB):
//   transformed : N*R*DO f32   = 204,800,000 B
//   cnt         : N int
//   offs        : (N+1) int
//   cur         : N int
//   epack       : E int        = 6.4 MB
//   enorm       : E f32        = 6.4 MB
// ---------------------------------------------------------------------------
extern "C" void kernel_launch(void* const* d_in, const int* in_sizes, int n_in,
                              void* d_out, int out_size, void* d_ws, size_t ws_size,
                              hipStream_t stream) {
    const float* h    = (const float*)d_in[0];
    const float* w    = (const float*)d_in[1];
    const float* nrm  = (const float*)d_in[2];
    const int*   esrc = (const int*)d_in[3];
    const int*   edst = (const int*)d_in[4];
    const int*   erel = (const int*)d_in[5];
    float*       out  = (float*)d_out;

    char*  ws  = (char*)d_ws;
    size_t off = 0;
    auto carve = [&](size_t bytes) -> void* {
        void* p = ws + off;
        off = (off + bytes + 255) & ~(size_t)255;
        return p;
    };
    float* t     = (float*)carve((size_t)NN * RR * DO_ * sizeof(float));
    int*   cnt   = (int*)  carve((size_t)NN * sizeof(int));
    int*   offs  = (int*)  carve((size_t)(NN + 1) * sizeof(int));
    int*   cur   = (int*)  carve((size_t)NN * sizeof(int));
    int*   epack = (int*)  carve((size_t)EE * sizeof(int));
    float* enorm = (float*)carve((size_t)EE * sizeof(float));
    (void)ws_size; (void)in_sizes; (void)n_in; (void)out_size;

    k_zero   <<<(NN + 255) / 256, 256, 0, stream>>>(cnt, NN);
    k_gemm   <<<NT, 256, 0, stream>>>(h, w, t);
    k_hist   <<<(EE + 255) / 256, 256, 0, stream>>>(edst, cnt);
    k_scan   <<<1, 1024, 0, stream>>>(cnt, offs, cur);
    k_scatter<<<(EE + 255) / 256, 256, 0, stream>>>(esrc, edst, erel, nrm,
                                                    cur, epack, enorm);
    k_gather <<<(NN / 8), 256, 0, stream>>>(t, offs, epack, enorm, out);
}